// MaxPooling_15994458210504
// MI455X (gfx1250) — compile-verified
//
#include <hip/hip_runtime.h>
#include <math.h>

// ---------------- problem constants (from reference) ----------------
#define HID        128
#define GRAPHS     1024
#define TPB        256
#define TILE_ROWS  64
#define ROW_PAD    132                      // floats per LDS row (pad vs bank conflicts)
#define CHUNKS_PER_ROW   32                 // 128 floats / 4 per b128 chunk
#define CHUNKS_PER_TILE  (TILE_ROWS * CHUNKS_PER_ROW)   // 2048
#define CHUNKS_PER_THREAD (CHUNKS_PER_TILE / TPB)       // 8
#define BUF_FLOATS (TILE_ROWS * ROW_PAD)    // 8448 floats per buffer

// CDNA5 async global->LDS mover (ASYNCcnt-tracked), per CDNA5 ISA / CDNA5 ISA.
__device__ __forceinline__ void async_ld_b128(unsigned lds_byte, unsigned long long gaddr) {
    asm volatile("global_load_async_to_lds_b128 %0, %1, off"
                 :: "v"(lds_byte), "v"(gaddr) : "memory");
}
#define WAIT_ASYNCCNT(n) asm volatile("s_wait_asynccnt %0" :: "i"(n) : "memory")

__device__ __forceinline__ int lower_bound_i32(const int* __restrict__ b, int n, int key) {
    int lo = 0, hi = n;
    while (lo < hi) { int m = (lo + hi) >> 1; if (b[m] < key) lo = m + 1; else hi = m; }
    return lo;
}

__global__ __launch_bounds__(TPB)
void graph_maxpool_kernel(const float* __restrict__ x,
                          const int*   __restrict__ batch,
                          float*       __restrict__ out,
                          int n_nodes) {
    // LDS: [ 2 x tile buffer | embedding row | per-row flags ]
    __shared__ __align__(16) float smem[2 * BUF_FLOATS + HID + TILE_ROWS];
    float*    embLds = smem + 2 * BUF_FLOATS;
    unsigned* flags  = (unsigned*)(embLds + HID);

    const int g   = blockIdx.x;
    const int tid = threadIdx.x;

    // sorted batch -> contiguous segment per graph
    const int start = lower_bound_i32(batch, n_nodes, g);
    const int end   = lower_bound_i32(batch, n_nodes, g + 1);
    const int rows  = end - start;

    float* emb_out = out;                               // [GRAPHS, HID]
    float* att_out = out + (size_t)GRAPHS * HID;        // [n_nodes]

    // -------- Phase 1: per-dim segment max (coalesced b128 sweep) --------
    const int c4   = tid & 31;       // float4 column 0..31
    const int part = tid >> 5;       // row octant 0..7
    float4 acc = make_float4(-INFINITY, -INFINITY, -INFINITY, -INFINITY);
    for (int r = start + part; r < end; r += 8) {
        const float4 v = *(const float4*)(x + (size_t)r * HID + c4 * 4);
        acc.x = fmaxf(acc.x, v.x); acc.y = fmaxf(acc.y, v.y);
        acc.z = fmaxf(acc.z, v.z); acc.w = fmaxf(acc.w, v.w);
    }
    float4* scr4 = (float4*)smem;        // reuse tile buffers as scratch (pre-staging)
    scr4[tid] = acc;
    __syncthreads();
    if (tid < 32) {
        float4 e = scr4[tid];
        #pragma unroll
        for (int p = 1; p < 8; ++p) {
            const float4 v = scr4[tid + 32 * p];
            e.x = fmaxf(e.x, v.x); e.y = fmaxf(e.y, v.y);
            e.z = fmaxf(e.z, v.z); e.w = fmaxf(e.w, v.w);
        }
        ((float4*)(emb_out + (size_t)g * HID))[tid] = e;     // segment_max output
        embLds[tid * 4 + 0] = e.x; embLds[tid * 4 + 1] = e.y;
        embLds[tid * 4 + 2] = e.z; embLds[tid * 4 + 3] = e.w;
    }
    if (tid < TILE_ROWS) flags[tid] = 0;
    __syncthreads();
    if (rows == 0) return;               // empty graph: emb = -inf, no nodes (uniform exit)

    // -------- Phase 2: is-max test via double-buffered async LDS staging --------
    const int nTiles = (rows + TILE_ROWS - 1) / TILE_ROWS;
    const unsigned           lds_base = (unsigned)(unsigned long long)(&smem[0]);
    const unsigned long long xg       = (unsigned long long)(size_t)x;

    auto stage = [&](int tile) {
        const unsigned bsel = (unsigned)(tile & 1);
        const int row0 = start + tile * TILE_ROWS;
        #pragma unroll
        for (int k = 0; k < CHUNKS_PER_THREAD; ++k) {
            const int j   = tid + k * TPB;          // 0..2047
            const int row = j >> 5;                 // 0..63
            const int c   = j & 31;                 // b128 chunk within row
            int grow = row0 + row;
            if (grow >= end) grow = end - 1;        // clamp: uniform ASYNCcnt per wave
            const unsigned long long ga =
                xg + (((unsigned long long)grow * HID + (unsigned)(c * 4)) * 4ull);
            const unsigned la =
                lds_base + bsel * (BUF_FLOATS * 4u) + (unsigned)row * (ROW_PAD * 4u) + (unsigned)c * 16u;
            async_ld_b128(la, ga);
        }
    };

    int cnt = 0;
    stage(0);
    for (int t = 0; t < nTiles; ++t) {
        if (t + 1 < nTiles) { stage(t + 1); WAIT_ASYNCCNT(CHUNKS_PER_THREAD); } // drain tile t only
        else                { WAIT_ASYNCCNT(0); }
        __syncthreads();

        const int rvalid = min(TILE_ROWS, rows - t * TILE_ROWS);
        const int row = tid & (TILE_ROWS - 1);
        const int q   = tid >> 6;                   // column quarter 0..3
        if (row < rvalid) {
            const float* bp = smem + (t & 1) * BUF_FLOATS + row * ROW_PAD + q * 32;
            const float* ep = embLds + q * 32;
            bool any = false;
            #pragma unroll
            for (int k = 0; k < 8; ++k) {
                const float4 v = *(const float4*)(bp + k * 4);
                const float4 e = *(const float4*)(ep + k * 4);
                any = any || (v.x == e.x) || (v.y == e.y) || (v.z == e.z) || (v.w == e.w);
            }
            if (any) atomicOr(&flags[row], 1u);
        }
        __syncthreads();
        if (tid < rvalid) {
            const int s = flags[tid] ? 1 : 0;
            att_out[start + t * TILE_ROWS + tid] = (float)s;   // binary score
            cnt += s;
            flags[tid] = 0;                                    // visible by next tile's barrier
        }
    }

    // -------- Phase 3: normalize scores within the graph --------
    __syncthreads();                       // buffers idle: reuse as int scratch
    int* scrI = (int*)smem;
    scrI[tid] = cnt;
    __syncthreads();
    #pragma unroll
    for (int s = TPB / 2; s > 0; s >>= 1) {
        if (tid < s) scrI[tid] += scrI[tid + s];
        __syncthreads();
    }
    const int total = scrI[0];             // >= 1 for any non-empty graph
    const float inv = 1.0f / (float)total;
    for (int i = start + tid; i < end; i += TPB) att_out[i] *= inv;
}

extern "C" void kernel_launch(void* const* d_in, const int* in_sizes, int n_in,
                              void* d_out, int out_size, void* d_ws, size_t ws_size,
                              hipStream_t stream) {
    const float* x     = (const float*)d_in[0];
    const int*   batch = (const int*)d_in[1];
    (void)d_in; (void)n_in; (void)d_ws; (void)ws_size; (void)out_size;
    const int n_nodes = in_sizes[1];
    graph_maxpool_kernel<<<dim3(GRAPHS), dim3(TPB), 0, stream>>>(
        x, batch, (float*)d_out, n_nodes);
}